// ConditionalEditModel_74887049773678
// MI455X (gfx1250) — compile-verified
//
#include <hip/hip_runtime.h>

#define NNODES 20000
#define NBATCH 2
#define NEDGE  320000
#define HD     128
#define NET    8
#define NLAYER 4
#define MTOT   (NNODES * NBATCH)   /* 40000, divisible by 64 */
#define LN_EPS 1e-5f

typedef float v2f __attribute__((ext_vector_type(2)));
typedef float v8f __attribute__((ext_vector_type(8)));

#define GEMM_SILU  1
#define GEMM_ACCUM 2
#define GEMM_BBIAS 4

#define ROWTILES 4                  /* 64 rows per block */
#define LDS_STRIDE 132              /* 128 + 4 pad vs bank conflicts */

__device__ __forceinline__ float wave_sum(float v) {
  v += __shfl_xor(v, 16, 32);
  v += __shfl_xor(v, 8, 32);
  v += __shfl_xor(v, 4, 32);
  v += __shfl_xor(v, 2, 32);
  v += __shfl_xor(v, 1, 32);
  return v;
}

// SiLU with hardware v_rcp_f32 (avoids IEEE div expansion in the epilogue)
__device__ __forceinline__ float silu_f(float x) {
  return x * __builtin_amdgcn_rcpf(1.0f + __expf(-x));
}

// ---------------------------------------------------------------------------
// Pack W (128x128 row-major, W[k*128+n]) into WMMA B-fragment order:
// for n-tile t (0..7), k-step s (0..31), lane l (0..31), j (0..1):
//   packed[(((t*32+s)*32)+l)*2+j] = W[(4s + 2*(l>>4) + j)*128 + 16t + (l&15)]
// After packing, each wave's B fragment for step s is one coalesced b64 load.
// ---------------------------------------------------------------------------
__global__ void packW_kernel(const float* __restrict__ W, float* __restrict__ Wp)
{
  int idx = blockIdx.x * 256 + threadIdx.x;    // 16384 total
  if (idx >= 8 * 32 * 32 * 2) return;
  int j = idx & 1;
  int l = (idx >> 1) & 31;
  int s = (idx >> 6) & 31;
  int t = idx >> 11;
  int k = 4 * s + 2 * (l >> 4) + j;
  int n = 16 * t + (l & 15);
  Wp[idx] = W[(size_t)k * HD + n];
}

// ---------------------------------------------------------------------------
// M x 128 x 128 GEMM using V_WMMA_F32_16X16X4_F32 (fp32, wave32).
// Block: 256 threads = 8 waves; block tile = 64 rows x 128 cols.
// Wave w owns n-tile w and 4 row-tiles (4 accumulators). Per k-step:
// one coalesced global b64 (packed B fragment) + LDS A reads + 4 WMMAs.
// ---------------------------------------------------------------------------
__global__ __launch_bounds__(256) void gemm128_wmma(
    const float* __restrict__ A, const float* __restrict__ Wp,
    const float* __restrict__ bias, float* __restrict__ out, int flags)
{
  __shared__ float As[64 * LDS_STRIDE];
  const int tid = threadIdx.x;
  const int m0  = blockIdx.x * 64;

  // stage 64 x 128 A tile
  for (int idx = tid; idx < 64 * HD; idx += 256) {
    int r = idx >> 7, c = idx & 127;
    As[r * LDS_STRIDE + c] = A[(size_t)(m0 + r) * HD + c];
  }
  __syncthreads();

  const int wave = tid >> 5;
  const int lane = tid & 31;
  const int half = lane >> 4;   // 0: lanes 0-15 (K lo pair), 1: lanes 16-31 (K hi pair)
  const int l15  = lane & 15;
  const int col  = wave * 16 + l15;

  v8f acc[ROWTILES];
  if (flags & GEMM_ACCUM) {
    #pragma unroll
    for (int r = 0; r < ROWTILES; ++r)
      #pragma unroll
      for (int rr = 0; rr < 8; ++rr)
        acc[r][rr] = out[(size_t)(m0 + 16 * r + 8 * half + rr) * HD + col];
  } else {
    #pragma unroll
    for (int r = 0; r < ROWTILES; ++r) {
      float bv = 0.0f;
      if (bias) {
        int bofs = (flags & GEMM_BBIAS) ? ((m0 + 16 * r) / NNODES) * HD : 0;
        bv = bias[bofs + col];
      }
      #pragma unroll
      for (int rr = 0; rr < 8; ++rr) acc[r][rr] = bv;
    }
  }

  const v2f* __restrict__ Wp2 = (const v2f*)Wp;
  const int wbase = wave * 32 * 32 + lane;     // (wave*32 + s)*32 + lane, s=0

  #pragma unroll 4
  for (int s = 0; s < 32; ++s) {               // k-step: K = 4*s
    v2f b = Wp2[wbase + s * 32];
    const int ka = 4 * s + 2 * half;
    #pragma unroll
    for (int r = 0; r < ROWTILES; ++r) {
      v2f a;
      a.x = As[(16 * r + l15) * LDS_STRIDE + ka];
      a.y = As[(16 * r + l15) * LDS_STRIDE + ka + 1];
      acc[r] = __builtin_amdgcn_wmma_f32_16x16x4_f32(false, a, false, b,
                                                     (short)0, acc[r], false, false);
    }
  }

  #pragma unroll
  for (int r = 0; r < ROWTILES; ++r)
    #pragma unroll
    for (int rr = 0; rr < 8; ++rr) {
      float v = acc[r][rr];
      if (flags & GEMM_SILU) v = silu_f(v);
      out[(size_t)(m0 + 16 * r + 8 * half + rr) * HD + col] = v;
    }
}

// ---------------------------------------------------------------------------
// Per-batch seed density (mean of seed_mask over nodes)
// ---------------------------------------------------------------------------
__global__ void density_kernel(const float* __restrict__ seed, float* __restrict__ dens)
{
  __shared__ float red[256];
  int b = blockIdx.x;
  float s = 0.f;
  for (int i = threadIdx.x; i < NNODES; i += 256) s += seed[(size_t)b * NNODES + i];
  red[threadIdx.x] = s;
  __syncthreads();
  for (int off = 128; off > 0; off >>= 1) {
    if (threadIdx.x < off) red[threadIdx.x] += red[threadIdx.x + off];
    __syncthreads();
  }
  if (threadIdx.x == 0) dens[b] = red[0] / (float)NNODES;
}

// ---------------------------------------------------------------------------
// Encoder stage 1: build 9-dim node features, apply enc1 (K=9) + SiLU
// ---------------------------------------------------------------------------
__global__ void enc1_kernel(const float* __restrict__ pf, const float* __restrict__ seed,
                            const float* __restrict__ dens, const float* __restrict__ w,
                            const float* __restrict__ bvec, float* __restrict__ out)
{
  size_t idx = (size_t)blockIdx.x * 256 + threadIdx.x;
  if (idx >= (size_t)MTOT * HD) return;
  int row = (int)(idx >> 7), n = (int)(idx & 127);
  int b = row / NNODES, i = row - b * NNODES;
  float s = seed[(size_t)b * NNODES + i];
  float d = dens[b];
  float x[9];
  x[0] = pf[i * 3 + 0]; x[1] = pf[i * 3 + 1]; x[2] = pf[i * 3 + 2];
  x[3] = s; x[4] = 1.f - s; x[5] = s - 0.5f;
  x[6] = d; x[7] = s * d;   x[8] = (s - 0.5f) * d;
  float acc = bvec[n];
  #pragma unroll
  for (int j = 0; j < 9; ++j) acc += x[j] * w[j * HD + n];
  out[idx] = silu_f(acc);
}

// ---------------------------------------------------------------------------
// Per-edge-type constants: Ke = emb@Wk (no bias), Ve = emb@Wv, ebias = emb@ebw+ebb
// ---------------------------------------------------------------------------
__global__ void edgeconst_kernel(const float* __restrict__ emb,
                                 const float* __restrict__ Wk, const float* __restrict__ Wv,
                                 const float* __restrict__ ebw, const float* __restrict__ ebb,
                                 float* __restrict__ Ke, float* __restrict__ Ve,
                                 float* __restrict__ ebias)
{
  for (int idx = threadIdx.x; idx < NET * HD; idx += 256) {
    int et = idx >> 7, n = idx & 127;
    const float* e = emb + (size_t)et * HD;
    float sk = 0.f, sv = 0.f;
    for (int j = 0; j < HD; ++j) {
      float x = e[j];
      sk += x * Wk[(size_t)j * HD + n];
      sv += x * Wv[(size_t)j * HD + n];
    }
    Ke[idx] = sk; Ve[idx] = sv;
  }
  for (int et = threadIdx.x; et < NET; et += 256) {
    const float* e = emb + (size_t)et * HD;
    float s = 0.f;
    for (int j = 0; j < HD; ++j) s += e[j] * ebw[j];
    ebias[et] = s + ebb[0];
  }
}

// ---------------------------------------------------------------------------
// Per-layer init: agg = 0, m = ordered(-inf), denom = 0
// ---------------------------------------------------------------------------
__global__ void layer_init_kernel(float* __restrict__ agg, float* __restrict__ mbuf,
                                  float* __restrict__ denom)
{
  size_t idx = (size_t)blockIdx.x * 256 + threadIdx.x;
  if (idx < (size_t)MTOT * HD) agg[idx] = 0.f;
  if (idx < MTOT) { ((int*)mbuf)[idx] = (int)0x80000000; denom[idx] = 0.f; }
}

// ---------------------------------------------------------------------------
// Edge pass A: logits + per-target max (ordered-int atomicMax). 1 wave / edge.
// ---------------------------------------------------------------------------
__global__ __launch_bounds__(256) void edge_logits_kernel(
    const int* __restrict__ ei, const int* __restrict__ etype,
    const float* __restrict__ Qn, const float* __restrict__ Kn,
    const float* __restrict__ Ke, const float* __restrict__ ebias,
    float* __restrict__ logits, float* __restrict__ mbuf)
{
  int gw = blockIdx.x * 8 + (threadIdx.x >> 5);
  if (gw >= NBATCH * NEDGE) return;
  int b = gw / NEDGE;
  int e = gw - b * NEDGE;
  int lane = threadIdx.x & 31;
  int src = ei[e], tgt = ei[NEDGE + e], et = etype[e];
  const float4* q  = (const float4*)(Qn + ((size_t)b * NNODES + tgt) * HD);
  const float4* k  = (const float4*)(Kn + ((size_t)b * NNODES + src) * HD);
  const float4* ke = (const float4*)(Ke + (size_t)et * HD);
  float4 qv = q[lane], kv = k[lane], kev = ke[lane];
  float d = qv.x * (kv.x + kev.x) + qv.y * (kv.y + kev.y) +
            qv.z * (kv.z + kev.z) + qv.w * (kv.w + kev.w);
  d = wave_sum(d);
  if (lane == 0) {
    float logit = d * 0.08838834764831845f + ebias[et];  // 1/sqrt(128)
    logits[(size_t)b * NEDGE + e] = logit;
    int enc = __float_as_int(logit);
    enc = enc >= 0 ? enc : (int)(enc ^ 0x7FFFFFFF);
    atomicMax((int*)mbuf + (size_t)b * NNODES + tgt, enc);
  }
}

// ---------------------------------------------------------------------------
// Edge pass B: ex = exp(logit - m[tgt]); scatter Σex and Σex*v. 1 wave / edge.
// ---------------------------------------------------------------------------
__global__ __launch_bounds__(256) void edge_scatter_kernel(
    const int* __restrict__ ei, const int* __restrict__ etype,
    const float* __restrict__ Vn, const float* __restrict__ Ve,
    const float* __restrict__ logits, const float* __restrict__ mbuf,
    float* __restrict__ denom, float* __restrict__ agg)
{
  int gw = blockIdx.x * 8 + (threadIdx.x >> 5);
  if (gw >= NBATCH * NEDGE) return;
  int b = gw / NEDGE;
  int e = gw - b * NEDGE;
  int lane = threadIdx.x & 31;
  int src = ei[e], tgt = ei[NEDGE + e], et = etype[e];
  float logit = logits[(size_t)b * NEDGE + e];
  int mi = ((const int*)mbuf)[(size_t)b * NNODES + tgt];
  float mv = __int_as_float(mi >= 0 ? mi : (int)(mi ^ 0x7FFFFFFF));
  float ex = __expf(logit - mv);
  if (lane == 0) atomicAdd(denom + (size_t)b * NNODES + tgt, ex);
  const float4* v  = (const float4*)(Vn + ((size_t)b * NNODES + src) * HD);
  const float4* ve = (const float4*)(Ve + (size_t)et * HD);
  float4 vv = v[lane], vev = ve[lane];
  float* dst = agg + ((size_t)b * NNODES + tgt) * HD + lane * 4;
  atomicAdd(dst + 0, ex * (vv.x + vev.x));
  atomicAdd(dst + 1, ex * (vv.y + vev.y));
  atomicAdd(dst + 2, ex * (vv.z + vev.z));
  atomicAdd(dst + 3, ex * (vv.w + vev.w));
}

// agg /= denom (isolated nodes -> 0)
__global__ void agg_norm_kernel(float* __restrict__ agg, const float* __restrict__ denom)
{
  size_t idx = (size_t)blockIdx.x * 256 + threadIdx.x;
  if (idx >= (size_t)MTOT * HD) return;
  float dn = denom[idx >> 7];
  agg[idx] = dn > 0.f ? agg[idx] / dn : 0.f;
}

// x = hidden + upd; LayerNorm; write back to hidden. 1 wave / node row.
__global__ __launch_bounds__(256) void resid_ln_kernel(
    float* __restrict__ hidden, const float* __restrict__ upd,
    const float* __restrict__ g, const float* __restrict__ bb)
{
  int row = blockIdx.x * 8 + (threadIdx.x >> 5);
  if (row >= MTOT) return;
  int lane = threadIdx.x & 31;
  const float4* hp = (const float4*)(hidden + (size_t)row * HD);
  const float4* up = (const float4*)(upd + (size_t)row * HD);
  float4 h = hp[lane], u = up[lane];
  float x0 = h.x + u.x, x1 = h.y + u.y, x2 = h.z + u.z, x3 = h.w + u.w;
  float mu = wave_sum(x0 + x1 + x2 + x3) * (1.f / HD);
  float d0 = x0 - mu, d1 = x1 - mu, d2 = x2 - mu, d3 = x3 - mu;
  float var = wave_sum(d0 * d0 + d1 * d1 + d2 * d2 + d3 * d3) * (1.f / HD);
  float inv = rsqrtf(var + LN_EPS);
  float4 gv = ((const float4*)g)[lane];
  float4 bv = ((const float4*)bb)[lane];
  float4 o;
  o.x = d0 * inv * gv.x + bv.x;
  o.y = d1 * inv * gv.y + bv.y;
  o.z = d2 * inv * gv.z + bv.z;
  o.w = d3 * inv * gv.w + bv.w;
  ((float4*)(hidden + (size_t)row * HD))[lane] = o;
}

// g[b][n] = mean over nodes of hidden. 1 block per (b, n).
__global__ void gmean_kernel(const float* __restrict__ hidden, float* __restrict__ g)
{
  __shared__ float red[256];
  int bn = blockIdx.x;             // 0..B*HD-1
  int b = bn >> 7, n = bn & 127;
  float s = 0.f;
  for (int i = threadIdx.x; i < NNODES; i += 256)
    s += hidden[((size_t)b * NNODES + i) * HD + n];
  red[threadIdx.x] = s;
  __syncthreads();
  for (int off = 128; off > 0; off >>= 1) {
    if (threadIdx.x < off) red[threadIdx.x] += red[threadIdx.x + off];
    __syncthreads();
  }
  if (threadIdx.x == 0) g[bn] = red[0] / (float)NNODES;
}

// gb[b][n] = g[b] @ ro1w[128:256] + ro1b  (per-batch bias vector for readout GEMM)
__global__ void gbias_kernel(const float* __restrict__ g, const float* __restrict__ ro1w,
                             const float* __restrict__ ro1b, float* __restrict__ gb)
{
  int idx = threadIdx.x;           // exactly B*HD = 256
  int b = idx >> 7, n = idx & 127;
  float s = ro1b[n];
  for (int j = 0; j < HD; ++j) s += g[b * HD + j] * ro1w[(size_t)(HD + j) * HD + n];
  gb[idx] = s;
}

// out[row] = r1[row] @ ro2w + ro2b. 1 wave / row.
__global__ __launch_bounds__(256) void final_kernel(
    const float* __restrict__ r1, const float* __restrict__ w,
    const float* __restrict__ b, float* __restrict__ out)
{
  int row = blockIdx.x * 8 + (threadIdx.x >> 5);
  if (row >= MTOT) return;
  int lane = threadIdx.x & 31;
  float4 rv = ((const float4*)(r1 + (size_t)row * HD))[lane];
  float4 wv = ((const float4*)w)[lane];
  float d = rv.x * wv.x + rv.y * wv.y + rv.z * wv.z + rv.w * wv.w;
  d = wave_sum(d);
  if (lane == 0) out[row] = d + b[0];
}

// ---------------------------------------------------------------------------
extern "C" void kernel_launch(void* const* d_in, const int* in_sizes, int n_in,
                              void* d_out, int out_size, void* d_ws, size_t ws_size,
                              hipStream_t stream)
{
  (void)in_sizes; (void)n_in; (void)out_size; (void)ws_size;

  const float* pf    = (const float*)d_in[0];
  const float* seed  = (const float*)d_in[1];
  const int*   ei    = (const int*)d_in[2];
  const int*   ety   = (const int*)d_in[3];
  const float* enc1b = (const float*)d_in[4];
  const float* enc1w = (const float*)d_in[5];
  const float* enc2b = (const float*)d_in[6];
  const float* enc2w = (const float*)d_in[7];
  const float* ro1b  = (const float*)d_in[68];
  const float* ro1w  = (const float*)d_in[69];
  const float* ro2b  = (const float*)d_in[70];
  const float* ro2w  = (const float*)d_in[71];

  float* ws = (float*)d_ws;
  const size_t MH = (size_t)MTOT * HD;
  float* hidden = ws;
  float* bufQ   = hidden + MH;
  float* bufK   = bufQ + MH;
  float* bufV   = bufK + MH;
  float* agg    = bufV + MH;
  float* logits = agg + MH;
  float* mbuf   = logits + (size_t)NBATCH * NEDGE;
  float* denom  = mbuf + MTOT;
  float* Ke     = denom + MTOT;
  float* Ve     = Ke + NET * HD;
  float* ebias  = Ve + NET * HD;
  float* dens   = ebias + NET;
  float* g      = dens + NBATCH;
  float* gb     = g + NBATCH * HD;
  float* Wp     = gb + NBATCH * HD;    // 16384 floats packed-W scratch

  dim3 blk(256);
  const int gemmGrid = MTOT / 64;                       // 625
  const int packGrid = (8 * 32 * 32 * 2) / 256;         // 64
  const int mhGrid   = (int)((MH + 255) / 256);         // 20000
  const int edgeGrid = (NBATCH * NEDGE) / 8;            // 80000
  const int rowGrid  = (MTOT + 7) / 8;                  // 5000

  #define RUN_GEMM(Aptr, Wptr, Bptr, Optr, Fl)                                   \
    do {                                                                         \
      packW_kernel<<<packGrid, blk, 0, stream>>>((Wptr), Wp);                    \
      gemm128_wmma<<<gemmGrid, blk, 0, stream>>>((Aptr), Wp, (Bptr), (Optr), (Fl)); \
    } while (0)

  // Encoder
  density_kernel<<<NBATCH, blk, 0, stream>>>(seed, dens);
  enc1_kernel<<<mhGrid, blk, 0, stream>>>(pf, seed, dens, enc1w, enc1b, bufQ);
  RUN_GEMM(bufQ, enc2w, enc2b, hidden, 0);

  // GNN layers
  for (int l = 0; l < NLAYER; ++l) {
    int base = 8 + 15 * l;
    const float* ebb = (const float*)d_in[base + 0];
    const float* ebw = (const float*)d_in[base + 1];
    const float* emb = (const float*)d_in[base + 2];
    const float* kb  = (const float*)d_in[base + 3];
    const float* kw  = (const float*)d_in[base + 4];
    const float* lnb = (const float*)d_in[base + 5];
    const float* lng = (const float*)d_in[base + 6];
    const float* qb  = (const float*)d_in[base + 7];
    const float* qw  = (const float*)d_in[base + 8];
    const float* u1b = (const float*)d_in[base + 9];
    const float* u1w = (const float*)d_in[base + 10];
    const float* u2b = (const float*)d_in[base + 11];
    const float* u2w = (const float*)d_in[base + 12];
    const float* vb  = (const float*)d_in[base + 13];
    const float* vw  = (const float*)d_in[base + 14];

    RUN_GEMM(hidden, qw, qb, bufQ, 0);
    RUN_GEMM(hidden, kw, kb, bufK, 0);
    RUN_GEMM(hidden, vw, vb, bufV, 0);
    edgeconst_kernel<<<1, blk, 0, stream>>>(emb, kw, vw, ebw, ebb, Ke, Ve, ebias);
    layer_init_kernel<<<mhGrid, blk, 0, stream>>>(agg, mbuf, denom);
    edge_logits_kernel<<<edgeGrid, blk, 0, stream>>>(ei, ety, bufQ, bufK, Ke, ebias,
                                                     logits, mbuf);
    edge_scatter_kernel<<<edgeGrid, blk, 0, stream>>>(ei, ety, bufV, Ve, logits, mbuf,
                                                      denom, agg);
    agg_norm_kernel<<<mhGrid, blk, 0, stream>>>(agg, denom);
    // upd = silu(hidden@W1a + agg@W1b + b1) @ W2 + b2  (u1 split + accumulate)
    RUN_GEMM(hidden, u1w, u1b, bufQ, 0);
    RUN_GEMM(agg, u1w + HD * HD, nullptr, bufQ, GEMM_ACCUM | GEMM_SILU);
    RUN_GEMM(bufQ, u2w, u2b, bufK, 0);
    resid_ln_kernel<<<rowGrid, blk, 0, stream>>>(hidden, bufK, lng, lnb);
  }

  // Readout
  gmean_kernel<<<NBATCH * HD, blk, 0, stream>>>(hidden, g);
  gbias_kernel<<<1, blk, 0, stream>>>(g, ro1w, ro1b, gb);
  RUN_GEMM(hidden, ro1w, gb, bufQ, GEMM_SILU | GEMM_BBIAS);
  final_kernel<<<rowGrid, blk, 0, stream>>>(bufQ, ro2w, ro2b, (float*)d_out);

  #undef RUN_GEMM
}